// GraphConvolution_901943132168
// MI455X (gfx1250) — compile-verified
//
#include <hip/hip_runtime.h>
#include <hip/hip_bf16.h>

#define D_FEAT 128
#define EPW 64          // edges staged per wave (full tiles)
#define WAVES_PER_BLOCK 8
#define PFD 8           // prefetch distance (edges)

typedef int v2i __attribute__((ext_vector_type(2)));
typedef __attribute__((address_space(1))) v2i g_v2i;
typedef __attribute__((address_space(3))) v2i l_v2i;

static __device__ __forceinline__ void wait_async0() {
#if __has_builtin(__builtin_amdgcn_s_wait_asynccnt)
  __builtin_amdgcn_s_wait_asynccnt(0);
#else
  asm volatile("s_wait_asynccnt 0" ::: "memory");
#endif
  __asm__ volatile("" ::: "memory");
}

static __device__ __forceinline__ void atomic_add_f32(float* p, float v) {
  // Non-returning hardware fp32 atomic (global_atomic_add_f32, STOREcnt).
  unsafeAtomicAdd(p, v);
}

// Shared per-edge body: endpoints are wave-uniform -> readfirstlane into
// SGPRs (SMEM norm loads, SADDR-form row addressing); one global_load_b128
// per lane (32 lanes x float4 = one 128-float row) and 4 non-returning fp32
// atomics per lane.
static __device__ __forceinline__ void gcn_edge(const float* __restrict__ feat,
                                                const float* __restrict__ norm_l,
                                                const float* __restrict__ norm_r,
                                                float* __restrict__ out,
                                                int s_in, int d_in, int lane) {
  const int s = __builtin_amdgcn_readfirstlane(s_in);
  const int d = __builtin_amdgcn_readfirstlane(d_in);
  const float w = norm_l[s] * norm_r[d];   // fused L/R normalization
  const float4 v =
      *(const float4*)(&feat[(size_t)s * D_FEAT + (size_t)lane * 4]);
  float* o = &out[(size_t)d * D_FEAT + (size_t)lane * 4];
  atomic_add_f32(o + 0, w * v.x);
  atomic_add_f32(o + 1, w * v.y);
  atomic_add_f32(o + 2, w * v.z);
  atomic_add_f32(o + 3, w * v.w);
}

// ---------------------------------------------------------------------------
// Hot kernel (first in file so the disasm snippet shows it): full EPW-edge
// tiles only. One wave owns EPW consecutive edges; indices staged into a
// wave-private LDS tile with gfx1250 async-to-LDS DMA (ASYNCcnt).
// ---------------------------------------------------------------------------
__global__ __launch_bounds__(WAVES_PER_BLOCK * 32)
void gcn_scatter_full_kernel(const float* __restrict__ feat,
                             const int* __restrict__ src,
                             const int* __restrict__ dst,
                             const float* __restrict__ norm_l,
                             const float* __restrict__ norm_r,
                             float* __restrict__ out,
                             int n_waves) {
  __shared__ int tile[WAVES_PER_BLOCK * 2 * EPW];

  const int lane = threadIdx.x & 31;
  const int wib  = threadIdx.x >> 5;
  const int wave_id = blockIdx.x * WAVES_PER_BLOCK + wib;
  if (wave_id >= n_waves) return;
  const long e0 = (long)wave_id * EPW;

  int* my = &tile[wib * 2 * EPW];   // [0..EPW): src, [EPW..2*EPW): dst

#if __has_builtin(__builtin_amdgcn_global_load_async_to_lds_b64)
  // 32 lanes x 8 B = 256 B = EPW ints per array.
  __builtin_amdgcn_global_load_async_to_lds_b64(
      (g_v2i*)(v2i*)(src + e0 + lane * 2),
      (l_v2i*)(v2i*)(my + lane * 2), 0, 0);
  __builtin_amdgcn_global_load_async_to_lds_b64(
      (g_v2i*)(v2i*)(dst + e0 + lane * 2),
      (l_v2i*)(v2i*)(my + EPW + lane * 2), 0, 0);
  wait_async0();
#else
  my[lane] = src[e0 + lane];
  my[lane + 32] = src[e0 + lane + 32];
  my[EPW + lane] = dst[e0 + lane];
  my[EPW + lane + 32] = dst[e0 + lane + 32];
#endif

#pragma unroll 4
  for (int i = 0; i < EPW - PFD; ++i) {
    // Pull a future source row toward the WGP while atomics drain.
    const int sp = __builtin_amdgcn_readfirstlane(my[i + PFD]);
    __builtin_prefetch(&feat[(size_t)sp * D_FEAT + (size_t)lane * 4], 0, 1);
    gcn_edge(feat, norm_l, norm_r, out, my[i], my[EPW + i], lane);
  }
#pragma unroll
  for (int i = EPW - PFD; i < EPW; ++i)
    gcn_edge(feat, norm_l, norm_r, out, my[i], my[EPW + i], lane);
}

// ---------------------------------------------------------------------------
// Tail kernel: remaining edges (< EPW), one wave per edge, no staging.
// ---------------------------------------------------------------------------
__global__ __launch_bounds__(WAVES_PER_BLOCK * 32)
void gcn_scatter_tail_kernel(const float* __restrict__ feat,
                             const int* __restrict__ src,
                             const int* __restrict__ dst,
                             const float* __restrict__ norm_l,
                             const float* __restrict__ norm_r,
                             float* __restrict__ out,
                             int e_begin, int n_edges) {
  const int lane = threadIdx.x & 31;
  const int e = e_begin + blockIdx.x * WAVES_PER_BLOCK + (threadIdx.x >> 5);
  if (e >= n_edges) return;
  gcn_edge(feat, norm_l, norm_r, out, src[e], dst[e], lane);
}

// ---------------------------------------------------------------------------
// K1: zero output (float4 stores -> global_store_b128) and degree scratch
// ---------------------------------------------------------------------------
__global__ __launch_bounds__(256) void gcn_zero_kernel(float4* __restrict__ out4,
                                                       long n4,
                                                       float* __restrict__ deg,
                                                       int ndeg) {
  const long i = (long)blockIdx.x * blockDim.x + threadIdx.x;
  const long stride = (long)gridDim.x * blockDim.x;
  const float4 z = make_float4(0.f, 0.f, 0.f, 0.f);
  for (long k = i; k < n4; k += stride) out4[k] = z;
  for (long k = i; k < ndeg; k += stride) deg[k] = 0.f;
}

// ---------------------------------------------------------------------------
// K2: degree counting (fp32 atomics; exact for counts < 2^24)
// ---------------------------------------------------------------------------
__global__ __launch_bounds__(256) void gcn_degree_kernel(const int* __restrict__ src,
                                                         const int* __restrict__ dst,
                                                         float* __restrict__ deg_o,
                                                         float* __restrict__ deg_i,
                                                         int n_edges) {
  const int i = blockIdx.x * blockDim.x + threadIdx.x;
  if (i < n_edges) {
    atomic_add_f32(&deg_o[src[i]], 1.0f);
    atomic_add_f32(&deg_i[dst[i]], 1.0f);
  }
}

// ---------------------------------------------------------------------------
// K3: deg -> rsqrt(max(deg,1)) in place (both halves in one pass)
// ---------------------------------------------------------------------------
__global__ __launch_bounds__(256) void gcn_norm_kernel(float* __restrict__ deg, int n) {
  const int i = blockIdx.x * blockDim.x + threadIdx.x;
  if (i < n) deg[i] = rsqrtf(fmaxf(deg[i], 1.0f));
}

// ---------------------------------------------------------------------------
extern "C" void kernel_launch(void* const* d_in, const int* in_sizes, int n_in,
                              void* d_out, int out_size, void* d_ws, size_t ws_size,
                              hipStream_t stream) {
  const float* feat = (const float*)d_in[0];
  const int*   src  = (const int*)d_in[1];
  const int*   dst  = (const int*)d_in[2];
  float* out = (float*)d_out;

  const int n_nodes = in_sizes[0] / D_FEAT;
  const int n_edges = in_sizes[1];

  float* deg_o = (float*)d_ws;          // becomes norm_l after K3
  float* deg_i = deg_o + n_nodes;       // becomes norm_r after K3

  const long n4 = (long)n_nodes * D_FEAT / 4;
  gcn_zero_kernel<<<2048, 256, 0, stream>>>((float4*)out, n4, deg_o, 2 * n_nodes);

  gcn_degree_kernel<<<(n_edges + 255) / 256, 256, 0, stream>>>(src, dst, deg_o,
                                                               deg_i, n_edges);

  gcn_norm_kernel<<<(2 * n_nodes + 255) / 256, 256, 0, stream>>>(deg_o,
                                                                 2 * n_nodes);

  const int n_full_waves = n_edges / EPW;
  if (n_full_waves > 0) {
    const int blocks = (n_full_waves + WAVES_PER_BLOCK - 1) / WAVES_PER_BLOCK;
    gcn_scatter_full_kernel<<<blocks, WAVES_PER_BLOCK * 32, 0, stream>>>(
        feat, src, dst, deg_o, deg_i, out, n_full_waves);
  }
  const int tail = n_edges - n_full_waves * EPW;
  if (tail > 0) {
    const int blocks = (tail + WAVES_PER_BLOCK - 1) / WAVES_PER_BLOCK;
    gcn_scatter_tail_kernel<<<blocks, WAVES_PER_BLOCK * 32, 0, stream>>>(
        feat, src, dst, deg_o, deg_i, out, n_full_waves * EPW, n_edges);
  }
}